// CTCBeamSearchDecoder_88390426952431
// MI455X (gfx1250) — compile-verified
//
#include <hip/hip_runtime.h>
#include <hip/hip_bf16.h>
#include <math.h>

#define B_ 64
#define T_ 1024
#define C_ 512
#define W_ 8
#define NEGF (-1e30f)

typedef float v2f __attribute__((ext_vector_type(2)));
typedef float v8f __attribute__((ext_vector_type(8)));

// ---------------------------------------------------------------------------
// Pass 1: per-row top-8 logits (value,index) and log-softmax base.
// One wave32 per row; 8 waves (256 threads) per block; B*T/8 = 8192 blocks.
// Row read is fully coalesced float4 loads. Sum-exp reduction uses
// v_wmma_f32_16x16x4_f32 with an all-ones B matrix: D[m,n] = sum_k A[m,k]+C,
// accumulated over 8 tiles (covers all 512 values), then per-lane sum of the
// 8 accumulator VGPRs + one shfl_xor(16) gives the full row sum in all lanes.
// ---------------------------------------------------------------------------
__global__ __launch_bounds__(256) void rowstats_kernel(
    const float* __restrict__ logits,
    float* __restrict__ topv,
    int*   __restrict__ topi)
{
    const int wave = threadIdx.x >> 5;       // 0..7
    const int lane = threadIdx.x & 31;       // wave32
    const int row  = blockIdx.x * 8 + wave;  // < B_*T_
    const float* __restrict__ src = logits + (size_t)row * C_;

    // Coalesced load: chunk c -> float4 at element c*128 + lane*4
    float e[16];
    int   ei[16];
#pragma unroll
    for (int c = 0; c < 4; ++c) {
        const float4 v = ((const float4*)src)[c * 32 + lane];
        const int base = c * 128 + lane * 4;
        e[c*4+0] = v.x; e[c*4+1] = v.y; e[c*4+2] = v.z; e[c*4+3] = v.w;
        ei[c*4+0] = base; ei[c*4+1] = base+1; ei[c*4+2] = base+2; ei[c*4+3] = base+3;
    }

    // Working copy for masked top-8 extraction (tie-break: lowest class index,
    // matching jax.lax.top_k's lowest-flat-index preference).
    float m[16];
#pragma unroll
    for (int i = 0; i < 16; ++i) m[i] = e[i];

    float tv[8]; int ti[8];
#pragma unroll
    for (int it = 0; it < 8; ++it) {
        float bv = m[0]; int bi = ei[0];
#pragma unroll
        for (int i = 1; i < 16; ++i) {
            const bool better = (m[i] > bv) || ((m[i] == bv) && (ei[i] < bi));
            bv = better ? m[i]  : bv;
            bi = better ? ei[i] : bi;
        }
#pragma unroll
        for (int off = 16; off > 0; off >>= 1) {
            const float ov = __shfl_xor(bv, off, 32);
            const int   oi = __shfl_xor(bi, off, 32);
            const bool better = (ov > bv) || ((ov == bv) && (oi < bi));
            bv = better ? ov : bv;
            bi = better ? oi : bi;
        }
        tv[it] = bv; ti[it] = bi;           // identical in all lanes
#pragma unroll
        for (int i = 0; i < 16; ++i) if (ei[i] == bi) m[i] = NEGF;
    }
    const float rowmax = tv[0];

    // exp(x - max); sum via WMMA (matrix pipe is free: pass is bandwidth-bound)
    float ex[16];
#pragma unroll
    for (int i = 0; i < 16; ++i) ex[i] = expf(e[i] - rowmax);

    v8f acc = {};
    const v2f ones = {1.0f, 1.0f};
#if __has_builtin(__builtin_amdgcn_wmma_f32_16x16x4_f32)
#pragma unroll
    for (int i = 0; i < 8; ++i) {
        v2f a = { ex[2*i], ex[2*i+1] };   // 64 values per WMMA across the wave
        acc = __builtin_amdgcn_wmma_f32_16x16x4_f32(
            /*neg_a=*/false, a, /*neg_b=*/false, ones,
            /*c_mod=*/(short)0, acc, /*reuse_a=*/false, /*reuse_b=*/false);
    }
    float partial = acc[0]+acc[1]+acc[2]+acc[3]+acc[4]+acc[5]+acc[6]+acc[7];
#else
    float partial = 0.f;
#pragma unroll
    for (int i = 0; i < 16; ++i) partial += ex[i];
#pragma unroll
    for (int off = 1; off < 16; off <<= 1) partial += __shfl_xor(partial, off, 32);
#endif
    const float total = partial + __shfl_xor(partial, 16, 32);
    const float lsbase = rowmax + logf(total);   // lp[c] = x[c] - lsbase

    // One store per lane r (static indices; values identical across lanes)
#pragma unroll
    for (int r = 0; r < 8; ++r) {
        if (lane == r) {
            topv[(size_t)row * 8 + r] = tv[r] - lsbase;
            topi[(size_t)row * 8 + r] = ti[r];
        }
    }
}

// ---------------------------------------------------------------------------
// Pass 2: sequential beam scan. One wave32 per batch element (64 blocks).
// 64 candidates/step (8 beams x top-8 classes); 8 butterfly-argmax picks with
// exact (sum desc, w asc, c asc) tie-break. History packed (cls<<3)|parent in
// LDS; backtrack + CTC collapse at the end. Next step's 64B register-
// prefetched to hide load latency in the serial loop.
// ---------------------------------------------------------------------------
__global__ __launch_bounds__(32) void beam_scan_kernel(
    const float* __restrict__ topv,
    const int*   __restrict__ topi,
    float* __restrict__ out_dec,
    float* __restrict__ out_len,
    float* __restrict__ out_score)
{
    __shared__ unsigned int hist[T_ * W_];   // (cls << 3) | parent
    __shared__ float s_lds[W_];
    __shared__ float ns_lds[W_];
    __shared__ float v_lds[W_];
    __shared__ int   i_lds[W_];
    __shared__ int   seq[T_];
    __shared__ int   sh_len;

    const int b    = blockIdx.x;
    const int lane = threadIdx.x;

    if (lane < W_) s_lds[lane] = (lane == 0) ? 0.0f : NEGF;
    __syncthreads();

    const int k1 = 2 * lane, k2 = 2 * lane + 1;   // my two of the 64 candidates
    const int w1 = k1 >> 3,  w2 = k2 >> 3;
    const int r1 = k1 & 7,   r2 = k2 & 7;

    // prefetch t = 0
    size_t base = (size_t)b * T_ * W_;
    float pv = topv[base + (lane & 7)];
    int   pi = topi[base + (lane & 7)];

    for (int t = 0; t < T_; ++t) {
        if (lane < 8) { v_lds[lane] = pv; i_lds[lane] = pi; }
        __syncthreads();
        if (t + 1 < T_) {                          // prefetch next step
            const size_t nb = ((size_t)b * T_ + (t + 1)) * W_;
            pv = topv[nb + (lane & 7)];
            pi = topi[nb + (lane & 7)];
        }
        float c1 = s_lds[w1] + v_lds[r1];
        float c2 = s_lds[w2] + v_lds[r2];

#pragma unroll
        for (int it = 0; it < 8; ++it) {
            // local best of my two (k1 < k2, so ties keep k1)
            const bool fb = (c1 >= c2);
            float bv = fb ? c1 : c2;
            int   bk = fb ? k1 : k2;
#pragma unroll
            for (int off = 16; off > 0; off >>= 1) {
                const float ov = __shfl_xor(bv, off, 32);
                const int   ok = __shfl_xor(bk, off, 32);
                const bool better = (ov > bv) || ((ov == bv) && (ok < bk));
                bv = better ? ov : bv;
                bk = better ? ok : bk;
            }
            if (lane == 0) {
                const int cls = i_lds[bk & 7];
                ns_lds[it] = bv;
                hist[t * W_ + it] = ((unsigned)cls << 3) | (unsigned)(bk >> 3);
            }
            if (bk == k1) c1 = NEGF;               // remove winner
            if (bk == k2) c2 = NEGF;
        }
        __syncthreads();
        if (lane < 8) s_lds[lane] = ns_lds[lane];
        __syncthreads();
    }

    // Backtrack best beam (index 0) and CTC-collapse.
    if (lane == 0) {
        int w = 0;
        for (int t = T_ - 1; t >= 0; --t) {
            const unsigned e = hist[t * W_ + w];
            seq[t] = (int)(e >> 3);
            w = (int)(e & 7u);
        }
        int prev = -1, len = 0;
        float* dec = out_dec + (size_t)b * T_;
        for (int t = 0; t < T_; ++t) {
            const int c = seq[t];
            if (c != 0) {                           // BLANK == 0
                if (c != prev) dec[len++] = (float)c;
                prev = c;
            }
        }
        sh_len = len;
        out_len[b]   = (float)len;
        out_score[b] = s_lds[0];
    }
    __syncthreads();
    const int len = sh_len;
    float* dec = out_dec + (size_t)b * T_;
    for (int i = len + lane; i < T_; i += 32) dec[i] = -1.0f;
}

extern "C" void kernel_launch(void* const* d_in, const int* in_sizes, int n_in,
                              void* d_out, int out_size, void* d_ws, size_t ws_size,
                              hipStream_t stream)
{
    (void)in_sizes; (void)n_in; (void)out_size; (void)ws_size;
    const float* logits = (const float*)d_in[0];

    float* topv = (float*)d_ws;                                  // B*T*8 floats (2MB)
    int*   topi = (int*)((char*)d_ws + (size_t)B_*T_*W_*sizeof(float)); // 2MB

    float* out       = (float*)d_out;
    float* out_dec   = out;                      // B*T
    float* out_len   = out + (size_t)B_ * T_;    // B
    float* out_score = out_len + B_;             // B

    rowstats_kernel<<<(B_ * T_) / 8, 256, 0, stream>>>(logits, topv, topi);
    beam_scan_kernel<<<B_, 32, 0, stream>>>(topv, topi, out_dec, out_len, out_score);
}